// InnerProduct_7868380086487
// MI455X (gfx1250) — compile-verified
//
#include <hip/hip_runtime.h>

typedef __attribute__((ext_vector_type(2))) float v2f;
typedef __attribute__((ext_vector_type(8))) float v8f;

#define BATCH   32768
#define NF      64
#define EMB     128
#define NOUT    10
#define ROWS    64      // batch rows per block
#define TPB     256     // 8 waves (wave32)

__global__ __launch_bounds__(TPB) void ip_fused_kernel(
    const float* __restrict__ x,      // [BATCH, NF, EMB]
    const float* __restrict__ w,      // [NOUT, NF]
    float* __restrict__ out)          // [BATCH, NOUT]
{
    __shared__ float xsq[ROWS][NF + 1];   // +1 pad: conflict-free column reads
    __shared__ float w2[NF * 16];         // w2[k*16 + n] = w[n,k]^2 (n<10 else 0)

    const int tid  = threadIdx.x;
    const int lane = tid & 31;
    const int wave = tid >> 5;
    const int r0   = blockIdx.x * ROWS;

    // ---- init w^2 tile in LDS (B-matrix layout: [K][N]) ----
    for (int i = tid; i < NF * 16; i += TPB) {
        int k = i >> 4, n = i & 15;
        float v = (n < NOUT) ? w[n * NF + k] : 0.0f;
        w2[i] = v * v;
    }

    // ---- phase 1: xsq[row][field] = sum_e x[r0+row, field, :]^2 ----
    // One (row,field) pair == 512B == 32 lanes x float4 -> perfectly coalesced.
    // Each wave takes 4 consecutive pairs per iter (4 B128 loads in flight).
    const float4* xv = (const float4*)x;
    const long pbase = (long)r0 * NF;
    for (int q = wave * 4; q < ROWS * NF; q += 32) {
        float s[4];
#pragma unroll
        for (int u = 0; u < 4; ++u) {
            float4 v = xv[(size_t)(pbase + q + u) * (EMB / 4) + lane];
            s[u] = v.x * v.x + v.y * v.y + v.z * v.z + v.w * v.w;
        }
#pragma unroll
        for (int u = 0; u < 4; ++u) {
            float t = s[u];
            t += __shfl_xor(t, 16, 32);
            t += __shfl_xor(t, 8, 32);
            t += __shfl_xor(t, 4, 32);
            t += __shfl_xor(t, 2, 32);
            t += __shfl_xor(t, 1, 32);
            s[u] = t;
        }
        if (lane == 0) {
#pragma unroll
            for (int u = 0; u < 4; ++u) {
                int qq = q + u;
                xsq[qq >> 6][qq & 63] = s[u];
            }
        }
    }
    __syncthreads();

    // ---- phase 2: ip[16-row tile, 16] = xsq[16x64] @ w2[64x16] via WMMA f32 ----
    // Waves 0..3 each own one 16-row tile; 16 chained V_WMMA_F32_16X16X4_F32.
    if (wave < 4) {
        const int row0 = wave * 16;
        const int half = lane >> 4;     // 0: K={0,1}, 1: K={2,3} within chunk
        const int lid  = lane & 15;     // A: M index, B: N index

        v8f acc = {};
#pragma unroll
        for (int kk = 0; kk < 16; ++kk) {
            const int k = kk * 4 + half * 2;
            v2f a, b;
            a.x = xsq[row0 + lid][k];
            a.y = xsq[row0 + lid][k + 1];
            b.x = w2[k * 16 + lid];
            b.y = w2[(k + 1) * 16 + lid];
            acc = __builtin_amdgcn_wmma_f32_16x16x4_f32(
                /*neg_a=*/false, a, /*neg_b=*/false, b,
                /*c_mod=*/(short)0, acc, /*reuse_a=*/false, /*reuse_b=*/false);
        }

        // D layout: VGPR j -> (M=j, N=lane) lanes 0-15 ; (M=j+8, N=lane-16) lanes 16-31
        if (lid < NOUT) {
#pragma unroll
            for (int j = 0; j < 8; ++j) {
                const int row = row0 + j + half * 8;
                out[(size_t)(r0 + row) * NOUT + lid] = acc[j];
            }
        }
    }
}

extern "C" void kernel_launch(void* const* d_in, const int* in_sizes, int n_in,
                              void* d_out, int out_size, void* d_ws, size_t ws_size,
                              hipStream_t stream) {
    const float* x = (const float*)d_in[0];   // [32768, 64, 128] f32
    const float* w = (const float*)d_in[1];   // [10, 64] f32
    float* out = (float*)d_out;               // [32768, 10] f32

    dim3 grid(BATCH / ROWS);   // 512 blocks
    dim3 block(TPB);
    ip_fused_kernel<<<grid, block, 0, stream>>>(x, w, out);
}